// TarNet_68719476736545
// MI455X (gfx1250) — compile-verified
//
#include <hip/hip_runtime.h>

typedef __attribute__((ext_vector_type(2))) float v2f;
typedef __attribute__((ext_vector_type(8))) float v8f;

#define WMMA_F32(a, b, c) \
  __builtin_amdgcn_wmma_f32_16x16x4_f32(false, (a), false, (b), (short)0, (c), false, false)

#if __has_builtin(__builtin_amdgcn_global_load_async_to_lds_b32) && \
    __has_builtin(__builtin_amdgcn_s_wait_asynccnt)
#define HAVE_ASYNC_LDS 1
#define TO_GLOBAL(p) ((__attribute__((address_space(1))) int*)(p))
#define TO_LDS(p)    ((__attribute__((address_space(3))) int*)(p))
#endif

// ---------------------------------------------------------------------------
// Kernel 0: zero the bucket counters
// ---------------------------------------------------------------------------
__global__ void zero_cnt_k(int* __restrict__ cnt) {
    if (threadIdx.x < 16) cnt[threadIdx.x] = 0;
}

// ---------------------------------------------------------------------------
// Kernel 1: bucket rows by cause (last column of input_mat)
// ---------------------------------------------------------------------------
__global__ void bucket_k(const float* __restrict__ input, int ldi,
                         int* __restrict__ cnt, int* __restrict__ list, int B) {
    int b = blockIdx.x * blockDim.x + threadIdx.x;
    if (b >= B) return;
    int t = (int)input[(size_t)b * ldi + (ldi - 1)];
    int pos = atomicAdd(&cnt[t], 1);
    list[t * B + pos] = b;
}

// ---------------------------------------------------------------------------
// Stage a 16(K) x 128(N) chunk of W into LDS, TRANSPOSED: dst[n][k] = W[kc+k][nT+n]
// 2048 elements, 8 per thread; global reads are 128-float coalesced row segments.
// Async path: DMA straight into the transposed LDS slot (per-lane LDS dest addr).
// ---------------------------------------------------------------------------
__device__ __forceinline__ void stage_w(const float* __restrict__ W, int N, int kc,
                                        int nTile, float (*__restrict__ dst)[18], int tid) {
    const int n  = tid & 127;
    const int k0 = tid >> 7;          // 0 or 1
    #pragma unroll
    for (int i = 0; i < 8; ++i) {
        const int k = k0 + 2 * i;
        const float* src = W + (size_t)(kc + k) * N + nTile + n;
#ifdef HAVE_ASYNC_LDS
        __builtin_amdgcn_global_load_async_to_lds_b32(TO_GLOBAL(src), TO_LDS(&dst[n][k]), 0, 0);
#else
        dst[n][k] = *src;
#endif
    }
}

__device__ __forceinline__ void stage_fence() {
#ifdef HAVE_ASYNC_LDS
    __builtin_amdgcn_s_wait_asynccnt(0);
#endif
    __syncthreads();
}

// ---------------------------------------------------------------------------
// 16-K chunk of WMMA MACs: B fragments come from transposed LDS as ds_load_b64.
// ---------------------------------------------------------------------------
__device__ __forceinline__ void mma_chunk(const float* __restrict__ Arow, int kcBase,
                                          const float (*__restrict__ wt)[18],
                                          int h, int ln, int nGrp,
                                          v8f& c0, v8f& c1, v8f& c2, v8f& c3) {
    #pragma unroll
    for (int kk = 0; kk < 16; kk += 4) {
        const int ka = kk + 2 * h;                       // K pair {0,1} or {2,3}
        v2f a = *(const v2f*)(Arow + kcBase + ka);
        const int col = nGrp * 64 + ln;
        v2f b0 = *(const v2f*)&wt[col +  0][ka];
        v2f b1 = *(const v2f*)&wt[col + 16][ka];
        v2f b2 = *(const v2f*)&wt[col + 32][ka];
        v2f b3 = *(const v2f*)&wt[col + 48][ka];
        c0 = WMMA_F32(a, b0, c0);
        c1 = WMMA_F32(a, b1, c1);
        c2 = WMMA_F32(a, b2, c2);
        c3 = WMMA_F32(a, b3, c3);
    }
}

// ---------------------------------------------------------------------------
// Kernel 2: C = relu(A @ W + bias).  Block tile 64(M) x 128(N), 8 waves,
// each wave 16x64 via 4 f32 WMMA accumulators. W double-buffered in LDS.
// ---------------------------------------------------------------------------
__global__ __launch_bounds__(256) void gemm_relu_k(
    const float* __restrict__ A, int lda,
    const float* __restrict__ W,      // K x N
    const float* __restrict__ bias,   // N
    float* __restrict__ C,            // M x N
    int N, int K)
{
    __shared__ __align__(16) float wt[2][128][18];   // [buf][n][k], padded rows

    const int tid  = threadIdx.x;
    const int wave = tid >> 5;
    const int lane = tid & 31;
    const int h    = lane >> 4;
    const int ln   = lane & 15;
    const int nGrp = wave >> 2;

    const int mTile  = blockIdx.x * 64 + (wave & 3) * 16;
    const int nBlock = blockIdx.y * 128;

    const float* Arow = A + (size_t)(mTile + ln) * lda;
    v8f c0 = {}, c1 = {}, c2 = {}, c3 = {};

    stage_w(W, N, 0, nBlock, wt[0], tid);
    stage_fence();

    const int NC = K >> 4;
    for (int c = 0; c < NC; ++c) {
        if (c + 1 < NC) stage_w(W, N, (c + 1) << 4, nBlock, wt[(c + 1) & 1], tid);
        mma_chunk(Arow, c << 4, wt[c & 1], h, ln, nGrp, c0, c1, c2, c3);
        stage_fence();                 // prefetch done + all reads of next buf retired
    }

    const int nTile = nBlock + nGrp * 64;
    const float bc0 = bias[nTile + 0  + ln];
    const float bc1 = bias[nTile + 16 + ln];
    const float bc2 = bias[nTile + 32 + ln];
    const float bc3 = bias[nTile + 48 + ln];
    #pragma unroll
    for (int j = 0; j < 8; ++j) {
        const int mg = mTile + j + 8 * h;                // D row = j + 8*half
        float* crow = C + (size_t)mg * N + nTile + ln;
        crow[0]  = fmaxf(c0[j] + bc0, 0.f);
        crow[16] = fmaxf(c1[j] + bc1, 0.f);
        crow[32] = fmaxf(c2[j] + bc2, 0.f);
        crow[48] = fmaxf(c3[j] + bc3, 0.f);
    }
}

// ---------------------------------------------------------------------------
// Kernel 3: per-head gathered GEMM + ReLU + dot(Wh2) + bh2  -> out[b]
// grid = (T, B/16); block = 256 (8 waves); wave w covers cols [64w, 64w+64).
// Waves split N disjointly -> B loads already coalesced & non-redundant.
// ---------------------------------------------------------------------------
__global__ __launch_bounds__(256) void head_k(
    const float* __restrict__ L,      // latent2: B x 512
    const float* __restrict__ Wh1,    // T x 512 x 512  (rows = contraction dim)
    const float* __restrict__ bh1,    // T x 512
    const float* __restrict__ Wh2,    // T x 512
    const float* __restrict__ bh2,    // T
    const int*   __restrict__ cnt,
    const int*   __restrict__ list,
    float* __restrict__ out, int B)
{
    const int H = 512;
    const int t = blockIdx.x;
    const int n = cnt[t];
    const int tile = blockIdx.y;
    if (tile * 16 >= n) return;       // uniform early-exit, EXEC stays full below

    const int tid  = threadIdx.x;
    const int wave = tid >> 5;
    const int lane = tid & 31;
    const int h    = lane >> 4;
    const int ln   = lane & 15;

    const int g = tile * 16 + ln;
    const int rowLn = (g < n) ? list[t * B + g] : list[t * B];  // pad w/ bucket row 0
    const float* Arow = L + (size_t)rowLn * H;
    const float* Wt   = Wh1 + (size_t)t * H * H;
    const int nBase = wave * 64;

    v8f c0 = {}, c1 = {}, c2 = {}, c3 = {};
    #pragma unroll 4
    for (int k = 0; k < H; k += 4) {
        const int ka = k + 2 * h;
        v2f a = *(const v2f*)(Arow + ka);
        const float* B0 = Wt + (size_t)ka * H + nBase + ln;
        const float* B1 = B0 + H;
        v2f b0 = { B0[0],  B1[0]  };
        v2f b1 = { B0[16], B1[16] };
        v2f b2 = { B0[32], B1[32] };
        v2f b3 = { B0[48], B1[48] };
        c0 = WMMA_F32(a, b0, c0);
        c1 = WMMA_F32(a, b1, c1);
        c2 = WMMA_F32(a, b2, c2);
        c3 = WMMA_F32(a, b3, c3);
    }

    // epilogue: relu(h + bh1) * Wh2, summed over this wave's 64 columns
    const float* b1t = bh1 + (size_t)t * H + nBase;
    const float* w2t = Wh2 + (size_t)t * H + nBase;
    float s[8];
    #pragma unroll
    for (int j = 0; j < 8; ++j) {
        float v0 = fmaxf(c0[j] + b1t[ln],      0.f) * w2t[ln];
        float v1 = fmaxf(c1[j] + b1t[16 + ln], 0.f) * w2t[16 + ln];
        float v2 = fmaxf(c2[j] + b1t[32 + ln], 0.f) * w2t[32 + ln];
        float v3 = fmaxf(c3[j] + b1t[48 + ln], 0.f) * w2t[48 + ln];
        s[j] = (v0 + v1) + (v2 + v3);
    }
    #pragma unroll
    for (int j = 0; j < 8; ++j) {     // reduce across the 16 lanes of each half
        s[j] += __shfl_xor(s[j], 1, 32);
        s[j] += __shfl_xor(s[j], 2, 32);
        s[j] += __shfl_xor(s[j], 4, 32);
        s[j] += __shfl_xor(s[j], 8, 32);
    }

    __shared__ float red[8][16];
    if (ln == 0) {
        #pragma unroll
        for (int j = 0; j < 8; ++j) red[wave][j + 8 * h] = s[j];
    }
    __syncthreads();

    if (tid < 16) {
        const int m = tid;
        float tot = 0.f;
        #pragma unroll
        for (int w = 0; w < 8; ++w) tot += red[w][m];
        const int gg = tile * 16 + m;
        if (gg < n) out[list[t * B + gg]] = tot + bh2[t];
    }
}

// ---------------------------------------------------------------------------
extern "C" void kernel_launch(void* const* d_in, const int* in_sizes, int n_in,
                              void* d_out, int out_size, void* d_ws, size_t ws_size,
                              hipStream_t stream) {
    const int B = 16384, H = 512, T = 16, LDI = 513;
    const float* input = (const float*)d_in[0];  // B x 513
    const float* W1    = (const float*)d_in[1];  // 512 x 512
    const float* b1    = (const float*)d_in[2];  // 512
    const float* W2    = (const float*)d_in[3];  // 512 x 512
    const float* b2    = (const float*)d_in[4];  // 512
    const float* Wh1   = (const float*)d_in[5];  // 16 x 512 x 512
    const float* bh1   = (const float*)d_in[6];  // 16 x 512
    const float* Wh2   = (const float*)d_in[7];  // 16 x 512
    const float* bh2   = (const float*)d_in[8];  // 16
    float* out = (float*)d_out;

    // workspace: latent1 (32MB) | latent2 (32MB) | cnt (32 ints) | list (T*B ints)
    float* lat1 = (float*)d_ws;
    float* lat2 = lat1 + (size_t)B * H;
    int*   cnt  = (int*)(lat2 + (size_t)B * H);
    int*   list = cnt + 32;

    zero_cnt_k<<<1, 32, 0, stream>>>(cnt);
    bucket_k<<<B / 256, 256, 0, stream>>>(input, LDI, cnt, list, B);

    dim3 gemmGrid(B / 64, H / 128);
    gemm_relu_k<<<gemmGrid, 256, 0, stream>>>(input, LDI, W1, b1, lat1, H, H);
    gemm_relu_k<<<gemmGrid, 256, 0, stream>>>(lat1,  H,   W2, b2, lat2, H, H);

    head_k<<<dim3(T, B / 16), 256, 0, stream>>>(lat2, Wh1, bh1, Wh2, bh2,
                                                cnt, list, out, B);
}